// HeadController_59356448031058
// MI455X (gfx1250) — compile-verified
//
#include <hip/hip_runtime.h>
#include <math.h>

#define Bn 64
#define Hn 4
#define Nn 16384
#define Wd 64
#define Dn 1024
#define Pn 70

typedef float v2f __attribute__((ext_vector_type(2)));
typedef float v8f __attribute__((ext_vector_type(8)));

// ---- workspace layout (floats) ----
// key   [B*H*64] @ 0
// knorm [256]    @ 16384
// beta  [256]    @ 16640
// gate  [256]    @ 16896
// s0    [256]    @ 17152
// s1    [256]    @ 17408
// s2    [256]    @ 17664
// gamma [256]    @ 17920
// ssum  [256]    @ 18176
#define WS_KEY   0
#define WS_KNORM 16384
#define WS_BETA  16640
#define WS_GATE  16896
#define WS_S0    17152
#define WS_S1    17408
#define WS_S2    17664
#define WS_GAM   17920
#define WS_SSUM  18176

static __device__ __forceinline__ float softplusf(float x) {
  return (x > 0.f) ? (x + log1pf(expf(-x))) : log1pf(expf(x));
}

// ---------------- Kernel A: read-head parameters ----------------
__global__ __launch_bounds__(256) void k_params(
    const float* __restrict__ cs, const float* __restrict__ Wr,
    const float* __restrict__ br, float* __restrict__ ws) {
  const int bh = blockIdx.x;           // 0..255
  const int b = bh >> 2, h = bh & 3;
  const int tid = threadIdx.x;
  const int lane = tid & 31, wave = tid >> 5;
  __shared__ float p[Pn];
  const float* csb = cs + (size_t)b * Dn;
  for (int j = wave; j < Pn; j += 8) {
    const float* wrow = Wr + (size_t)(h * Pn + j) * Dn;
    float acc = 0.f;
    for (int d = lane; d < Dn; d += 32) acc += csb[d] * wrow[d];
    #pragma unroll
    for (int off = 16; off; off >>= 1) acc += __shfl_down(acc, off);
    if (lane == 0) p[j] = acc + br[h * Pn + j];
  }
  __syncthreads();
  if (tid < Wd) ws[WS_KEY + bh * Wd + tid] = p[tid];
  if (tid == 0) {
    float sq = 0.f;
    for (int w = 0; w < Wd; ++w) sq += p[w] * p[w];
    ws[WS_KNORM + bh] = fmaxf(sqrtf(sq), 1e-8f);
    ws[WS_BETA  + bh] = softplusf(p[Wd]);
    ws[WS_GATE  + bh] = 1.f / (1.f + expf(-p[Wd + 1]));
    float m = fmaxf(p[Wd + 2], fmaxf(p[Wd + 3], p[Wd + 4]));
    float e0 = expf(p[Wd + 2] - m), e1 = expf(p[Wd + 3] - m), e2 = expf(p[Wd + 4] - m);
    float inv = 1.f / (e0 + e1 + e2);
    ws[WS_S0 + bh] = e0 * inv;
    ws[WS_S1 + bh] = e1 * inv;
    ws[WS_S2 + bh] = e2 * inv;
    ws[WS_GAM + bh] = 1.f + softplusf(p[Wd + 5]);
  }
}

// ------------- Kernel B: WMMA cosine-similarity logits -------------
// grid = B * 16 blocks of 256 threads (8 waves); each wave owns 16-row tiles.
// A (16x4 f32): lane m = lane&15 (heads 0..3 live, rest 0), VGPR pair = K
//   {0,1} for lanes 0..15, {2,3} for lanes 16..31, stepping w = 4k + ...
// B (4x16 f32): lane n = lane&15 (memory row in tile), same K striping.
// C (16x16 f32): c[h], lanes 0..15 hold dots[head=h][row=lane].
__global__ __launch_bounds__(256) void k_dots(
    const float* __restrict__ mem, const float* __restrict__ ws,
    float* __restrict__ logits /* d_out reused: [B][H][N] */) {
  const int blk = blockIdx.x;          // 0..1023
  const int b = blk >> 4;
  const int chunk = blk & 15;          // 1024 rows per chunk
  const int tid = threadIdx.x;
  const int lane = tid & 31, wave = tid >> 5;
  const int half = lane >> 4, r = lane & 15;

  v2f a[16];
  if (r < Hn) {
    const float* kp = ws + WS_KEY + (b * Hn + r) * Wd + 2 * half;
    #pragma unroll
    for (int k = 0; k < 16; ++k) { a[k].x = kp[4 * k]; a[k].y = kp[4 * k + 1]; }
  } else {
    #pragma unroll
    for (int k = 0; k < 16; ++k) { a[k].x = 0.f; a[k].y = 0.f; }
  }
  float kn[Hn], bt[Hn];
  #pragma unroll
  for (int h = 0; h < Hn; ++h) {
    kn[h] = ws[WS_KNORM + b * Hn + h];
    bt[h] = ws[WS_BETA  + b * Hn + h];
  }

  const int chunkStart = chunk * 1024;
  for (int it = 0; it < 8; ++it) {
    const int n0 = chunkStart + (it * 8 + wave) * 16;
    const float* mrow = mem + ((size_t)b * Nn + n0 + r) * Wd + 2 * half;
    v8f c = {};
    float sq = 0.f;
    #pragma unroll
    for (int k = 0; k < 16; ++k) {
      v2f bv = __builtin_nontemporal_load(
          reinterpret_cast<const v2f*>(mrow + 4 * k));
      sq += bv.x * bv.x + bv.y * bv.y;
      c = __builtin_amdgcn_wmma_f32_16x16x4_f32(
          false, a[k], false, bv, (short)0, c, false, false);
    }
    sq += __shfl_xor(sq, 16);            // combine lane halves (w%4 in {0,1} / {2,3})
    if (half == 0) {
      float mn = fmaxf(sqrtf(sq), 1e-8f);
      const int n = n0 + r;
      #pragma unroll
      for (int h = 0; h < Hn; ++h)
        logits[((size_t)(b * Hn + h)) * Nn + n] = c[h] * bt[h] / (kn[h] * mn);
    }
  }
}

// ------------- Kernel C: per-(b,h) softmax + gate + shift + sharpen -------------
__device__ __forceinline__ float blk_red(float v, bool is_sum, float* scratch) {
  const int lane = threadIdx.x & 31, wave = threadIdx.x >> 5;
  #pragma unroll
  for (int off = 16; off; off >>= 1) {
    float o = __shfl_down(v, off);
    v = is_sum ? (v + o) : fmaxf(v, o);
  }
  __syncthreads();                 // prior users of scratch region are done
  if (lane == 0) scratch[wave] = v;
  __syncthreads();
  float r = scratch[0];
  #pragma unroll
  for (int w = 1; w < 8; ++w) r = is_sum ? (r + scratch[w]) : fmaxf(r, scratch[w]);
  __syncthreads();
  return r;
}

__global__ __launch_bounds__(256) void k_row(
    const float* __restrict__ prevw, float* __restrict__ ws,
    float* __restrict__ out /* in: logits, out: unnormalized sharp */) {
  const int bh = blockIdx.x;           // 0..255
  const int tid = threadIdx.x;
  __shared__ float rowS[Nn];           // 64 KB; rowS[0..7] doubles as red scratch
  const float g  = ws[WS_GATE + bh];
  const float s0 = ws[WS_S0 + bh];
  const float s1 = ws[WS_S1 + bh];
  const float s2 = ws[WS_S2 + bh];
  const float gam = ws[WS_GAM + bh];
  const float* lrow = out + (size_t)bh * Nn;

  float m = -INFINITY;
  for (int n = tid; n < Nn; n += 256) m = fmaxf(m, lrow[n]);
  m = blk_red(m, false, rowS);

  float s = 0.f;
  for (int n = tid; n < Nn; n += 256) s += expf(lrow[n] - m);
  s = blk_red(s, true, rowS);
  const float inv = 1.f / s;

  const float* prow = prevw + (size_t)bh * Nn;
  for (int n = tid; n < Nn; n += 256) {
    float cw = expf(lrow[n] - m) * inv;
    rowS[n] = g * cw + (1.f - g) * prow[n];
  }
  __syncthreads();

  float acc = 0.f;
  float* orow = out + (size_t)bh * Nn;
  for (int n = tid; n < Nn; n += 256) {
    const int nm = (n == 0) ? (Nn - 1) : (n - 1);
    const int np = (n == Nn - 1) ? 0 : (n + 1);
    float sh = s0 * rowS[nm] + s1 * rowS[n] + s2 * rowS[np];
    float t = powf(sh, gam);
    acc += t;
    orow[n] = t;
  }
  acc = blk_red(acc, true, rowS);      // syncs before reusing rowS[0..7]
  if (tid == 0) ws[WS_SSUM + bh] = acc;
}

// ------------- Kernel D: final normalization -------------
__global__ __launch_bounds__(256) void k_norm(
    float* __restrict__ out, const float* __restrict__ ws) {
  const size_t i = (size_t)blockIdx.x * 256 + threadIdx.x;
  const int bh = (int)(i >> 14);       // N = 16384
  out[i] = out[i] / (ws[WS_SSUM + bh] + 1e-6f);
}

extern "C" void kernel_launch(void* const* d_in, const int* in_sizes, int n_in,
                              void* d_out, int out_size, void* d_ws, size_t ws_size,
                              hipStream_t stream) {
  const float* cs    = (const float*)d_in[0];   // (B, D)
  const float* prevw = (const float*)d_in[1];   // (B, H, N)
  const float* mem   = (const float*)d_in[2];   // (B, N, W)
  const float* Wr    = (const float*)d_in[3];   // (P*H, D)
  const float* br    = (const float*)d_in[4];   // (P*H,)
  float* out = (float*)d_out;                   // (B, H, N)
  float* ws  = (float*)d_ws;

  k_params<<<Bn * Hn, 256, 0, stream>>>(cs, Wr, br, ws);
  k_dots  <<<Bn * 16, 256, 0, stream>>>(mem, ws, out);
  k_row   <<<Bn * Hn, 256, 0, stream>>>(prevw, ws, out);
  k_norm  <<<(Bn * Hn * Nn) / 256, 256, 0, stream>>>(out, ws);
}